// CompressedSensing_14585708937842
// MI455X (gfx1250) — compile-verified
//
#include <hip/hip_runtime.h>
#include <cmath>

// ---------------- problem constants ----------------
#define MDIM   81       // measurement size (9*9)
#define MPAD   96       // 81 padded to 6*16
#define NDIM   5329     // sparse grid (73*73)
#define NPAD   5344     // 5329 padded to 334*16
#define NPAD2  5376     // row stride for y/ylast = KSPLIT*KCHUNK (uniform K splits)
#define BATCH  512
#define NITERS 100
#define KSPLIT 16
#define KCHUNK 336      // 16*336 = 5376
#define NT     (NPAD / 16)   // 334 n-column tiles
#define K4G2   (MPAD / 4)    // 24 WMMA k-steps in update GEMM

typedef __attribute__((ext_vector_type(2))) float v2f;
typedef __attribute__((ext_vector_type(8))) float v8f;

// ============ one-time repack: B operands in exact WMMA lane layout ============
// matW  [6][NPAD2/4][32][2] : B for G1 (y @ mat). k-dim = n, col = m-tile (6 tiles)
//   element (t,k4,l,v) = mat[n=k4*4+2*(l>>4)+v][m=t*16+(l&15)]  (0 padded)
__global__ void pack_w1_kernel(const float* __restrict__ mat, float* __restrict__ matW) {
  int i = blockIdx.x * blockDim.x + threadIdx.x;
  const int per_t = (NPAD2 / 4) * 64;
  if (i >= 6 * per_t) return;
  int t = i / per_t, rem = i % per_t;
  int k4 = rem / 64, q = rem % 64;
  int l = q >> 1, v = q & 1;
  int n = k4 * 4 + 2 * (l >> 4) + v;
  int m = t * 16 + (l & 15);
  matW[i] = (n < NDIM && m < MDIM) ? mat[n * MDIM + m] : 0.0f;
}
// matWT [334][24][32][2] : B for G2 (r @ mat^T). k-dim = m, col = n-tile
__global__ void pack_w2_kernel(const float* __restrict__ mat, float* __restrict__ matWT) {
  int i = blockIdx.x * blockDim.x + threadIdx.x;
  const int per_t = K4G2 * 64;
  if (i >= NT * per_t) return;
  int ct = i / per_t, rem = i % per_t;
  int k4 = rem / 64, q = rem % 64;
  int l = q >> 1, v = q & 1;
  int m = k4 * 4 + 2 * (l >> 4) + v;
  int n = ct * 16 + (l & 15);
  matWT[i] = (m < MDIM && n < NDIM) ? mat[n * MDIM + m] : 0.0f;
}

__global__ void zero_kernel(float* __restrict__ p, int n) {
  int i = blockIdx.x * blockDim.x + threadIdx.x;
  if (i < n) p[i] = 0.0f;
}

// ============ G1: rpart[ks] = y(32b x 336) @ mat-slice(336 x 96) ============
// block = 192 threads = 6 waves (one per m column tile), each wave keeps TWO
// accumulators (rows 0-15 / 16-31) sharing every B load. grid = (B/32, KSPLIT)
__global__ __launch_bounds__(192)
void fista_gemm1_kernel(const float* __restrict__ y,      // [BATCH][NPAD2]
                        const float* __restrict__ matW,   // repacked B
                        float* __restrict__ rpart) {      // [KSPLIT][BATCH][MPAD]
  __shared__ __align__(16) float ytile[32 * KCHUNK];      // 43008 B

  const int tid  = threadIdx.x;
  const int wave = tid >> 5;
  const int lane = tid & 31;
  const int row0 = blockIdx.x * 32;
  const int kbeg = blockIdx.y * KCHUNK;

  // cooperative coalesced load: 32 x 336 floats = 2688 float4, exactly 14/thread
#pragma unroll
  for (int j = 0; j < 14; ++j) {
    int i  = tid + j * 192;
    int rr = i / (KCHUNK / 4), cc = i % (KCHUNK / 4);
    *((float4*)(ytile + rr * KCHUNK) + cc) =
        *((const float4*)(y + (size_t)(row0 + rr) * NPAD2 + kbeg) + cc);
  }
  __syncthreads();

  const int hi = lane >> 4, lm = lane & 15;
  const float* asrc0 = ytile + lm * KCHUNK + 2 * hi;            // rows 0-15
  const float* asrc1 = asrc0 + 16 * KCHUNK;                     // rows 16-31
  const float* bsrc  = matW + ((size_t)wave * (NPAD2 / 4) + (kbeg >> 2)) * 64
                            + lane * 2;                         // +256B/step

  v8f acc0 = {}, acc1 = {};
#pragma unroll 8
  for (int k4 = 0; k4 < KCHUNK / 4; ++k4) {
    v2f b  = *(const v2f*)(bsrc + k4 * 64);    // one coalesced b64, reused twice
    v2f a0 = *(const v2f*)(asrc0 + k4 * 4);
    v2f a1 = *(const v2f*)(asrc1 + k4 * 4);
    acc0 = __builtin_amdgcn_wmma_f32_16x16x4_f32(
        false, a0, false, b, (short)0, acc0, false, false);
    acc1 = __builtin_amdgcn_wmma_f32_16x16x4_f32(
        false, a1, false, b, (short)0, acc1, false, false);
  }

  float* dst = rpart + ((size_t)blockIdx.y * BATCH + row0) * MPAD + wave * 16 + lm;
#pragma unroll
  for (int v = 0; v < 8; ++v) {
    dst[(size_t)(v + 8 * hi) * MPAD]      = acc0[v];
    dst[(size_t)(16 + v + 8 * hi) * MPAD] = acc1[v];
  }
}

// ============ residual: r = xc - sum_ks rpart (padded cols -> 0) ============
__global__ void fista_residual_kernel(const float* __restrict__ rpart,
                                      const float* __restrict__ x,   // (B,81,3)
                                      float* __restrict__ r,         // [BATCH][MPAD]
                                      int chan) {
  int i = blockIdx.x * blockDim.x + threadIdx.x;
  if (i >= BATCH * MPAD) return;
  int b = i / MPAD, m = i % MPAD;
  float s = 0.0f;
#pragma unroll
  for (int ks = 0; ks < KSPLIT; ++ks)
    s += rpart[((size_t)ks * BATCH + b) * MPAD + m];
  r[i] = (m < MDIM) ? (x[(size_t)(b * MDIM + m) * 3 + chan] - s) : 0.0f;
}

// ============ G2 + fused FISTA update ============
// block = 128 threads = 4 waves sharing one 32x96 r tile; each wave: one n-tile,
// TWO row-tile accumulators sharing every B load. grid = (B/32, ceil(334/4))
__global__ __launch_bounds__(128)
void fista_update_kernel(const float* __restrict__ r,      // [BATCH][MPAD]
                         const float* __restrict__ matWT,  // repacked B
                         float* __restrict__ y,            // [BATCH][NPAD2] (y_m)
                         float* __restrict__ ylast,        // [BATCH][NPAD2] (y_new)
                         const float* __restrict__ lamp,   // scalar
                         float coef,                       // (t-1)/t_next
                         float* __restrict__ out,          // (B,73,73,3)
                         int chan, int final_it) {
  __shared__ __align__(16) float rtile[32 * MPAD];         // 12288 B

  const int tid  = threadIdx.x;
  const int wave = tid >> 5;
  const int lane = tid & 31;
  const int row0 = blockIdx.x * 32;

  // cooperative load of r tile: 768 float4 over 128 threads, exactly 6 each
#pragma unroll
  for (int j = 0; j < 6; ++j) {
    int i  = tid + j * 128;
    int rr = i / (MPAD / 4), cc = i % (MPAD / 4);
    *((float4*)(rtile + rr * MPAD) + cc) =
        *((const float4*)(r + (size_t)(row0 + rr) * MPAD) + cc);
  }
  __syncthreads();

  const int colTile = blockIdx.y * 4 + wave;
  if (colTile >= NT) return;              // wave-uniform exit, no further barriers
  const int hi = lane >> 4, lm = lane & 15;

  const float* asrc0 = rtile + lm * MPAD + 2 * hi;
  const float* asrc1 = asrc0 + 16 * MPAD;
  const float* bsrc  = matWT + (size_t)colTile * (K4G2 * 64) + lane * 2;

  v8f acc0 = {}, acc1 = {};
#pragma unroll
  for (int k4 = 0; k4 < K4G2; ++k4) {
    v2f b  = *(const v2f*)(bsrc + k4 * 64);
    v2f a0 = *(const v2f*)(asrc0 + k4 * 4);
    v2f a1 = *(const v2f*)(asrc1 + k4 * 4);
    acc0 = __builtin_amdgcn_wmma_f32_16x16x4_f32(
        false, a0, false, b, (short)0, acc0, false, false);
    acc1 = __builtin_amdgcn_wmma_f32_16x16x4_f32(
        false, a1, false, b, (short)0, acc1, false, false);
  }

  const float lam = lamp[0];
  const int n_i = colTile * 16 + lm;
#pragma unroll
  for (int half = 0; half < 2; ++half) {
    const v8f acc = half ? acc1 : acc0;
#pragma unroll
    for (int v = 0; v < 8; ++v) {
      const int b_i = row0 + 16 * half + v + 8 * hi;
      const size_t idx = (size_t)b_i * NPAD2 + n_i;
      float yh   = y[idx] + acc[v];
      float ynew = copysignf(fmaxf(fabsf(yh) - lam, 0.0f), yh);  // soft threshold
      float ym   = ynew + coef * (ynew - ylast[idx]);            // momentum
      y[idx]     = ym;
      ylast[idx] = ynew;
      if (final_it && n_i < NDIM)
        out[((size_t)b_i * NDIM + n_i) * 3 + chan] = ynew;       // pre-momentum output
    }
  }
}

// ---------------- host orchestration ----------------
extern "C" void kernel_launch(void* const* d_in, const int* in_sizes, int n_in,
                              void* d_out, int out_size, void* d_ws, size_t ws_size,
                              hipStream_t stream) {
  const float* x   = (const float*)d_in[0];   // (512,9,9,3)
  const float* mat = (const float*)d_in[1];   // (5329,81)
  const float* lam = (const float*)d_in[2];   // scalar
  float* out = (float*)d_out;                 // (512,73,73,3)

  float* ws    = (float*)d_ws;
  float* y     = ws;                                        // B*NPAD2
  float* ylast = y     + (size_t)BATCH * NPAD2;             // B*NPAD2
  float* r     = ylast + (size_t)BATCH * NPAD2;             // B*MPAD
  float* rpart = r     + (size_t)BATCH * MPAD;              // KSPLIT*B*MPAD
  float* matW  = rpart + (size_t)KSPLIT * BATCH * MPAD;     // 6*(NPAD2/4)*64
  float* matWT = matW  + (size_t)6 * (NPAD2 / 4) * 64;      // NT*K4G2*64

  {
    int n1 = 6 * (NPAD2 / 4) * 64;
    pack_w1_kernel<<<(n1 + 255) / 256, 256, 0, stream>>>(mat, matW);
    int n2 = NT * K4G2 * 64;
    pack_w2_kernel<<<(n2 + 255) / 256, 256, 0, stream>>>(mat, matWT);
  }

  float t = 1.0f;  // momentum scalar persists across channels
  for (int c = 0; c < 3; ++c) {
    int nz = 2 * BATCH * NPAD2;  // y and ylast are contiguous
    zero_kernel<<<(nz + 255) / 256, 256, 0, stream>>>(y, nz);

    for (int it = 0; it < NITERS; ++it) {
      float tn   = (1.0f + sqrtf(1.0f + 4.0f * t * t)) * 0.5f;
      float coef = (t - 1.0f) / tn;
      t = tn;

      fista_gemm1_kernel<<<dim3(BATCH / 32, KSPLIT), 192, 0, stream>>>(y, matW, rpart);

      int nr = BATCH * MPAD;
      fista_residual_kernel<<<(nr + 255) / 256, 256, 0, stream>>>(rpart, x, r, c);

      fista_update_kernel<<<dim3(BATCH / 32, (NT + 3) / 4), 128, 0, stream>>>(
          r, matWT, y, ylast, lam, coef, out, c, (it == NITERS - 1) ? 1 : 0);
    }
  }
}